// sequenceMixer_67989332295813
// MI455X (gfx1250) — compile-verified
//
#include <hip/hip_runtime.h>
#include <cmath>

// ---------------- problem constants ----------------
#define B_    8
#define DIM_  1024
#define L_    4096
#define H_    512
#define EPS_  1e-4f

// ---------------- GEMM tiling ----------------
#define BM 128
#define BN 128
#define BK 32
#define APAD 36   // row stride of A tile in floats: 144B = 16B-aligned, bank-skewed
#define BPAD 132  // row stride of B tile in floats: 528B = 16B-aligned, bank-skewed

// ---------------- scan chunking ----------------
#define NROWS 8192      // B * 2 * H sequences
#define NCH   32        // chunks per sequence
#define CH    128       // chunk length (NCH*CH == L_)

typedef float v2f __attribute__((ext_vector_type(2)));
typedef float v8f __attribute__((ext_vector_type(8)));

// =====================================================================
// Kernel 1: EDM2 magnitude-preserving weight normalization.
// rows 0..1023 -> w_fore, 1024..2047 -> w_back (into wn[2048][1024]),
// rows 2048..3071 -> w_out (into won[1024][1024]).
// w_n = w / (EPS + ||w||/sqrt(1024)) / sqrt(1024),  sqrt(1024)=32.
// =====================================================================
__global__ __launch_bounds__(256) void norm_rows_kernel(
    const float* __restrict__ wf, const float* __restrict__ wb,
    const float* __restrict__ wo, float* __restrict__ wn,
    float* __restrict__ won) {
  int row = blockIdx.x;
  int t = threadIdx.x;
  const float* src;
  float* dst;
  if (row < 1024) {
    src = wf + (size_t)row * DIM_;
    dst = wn + (size_t)row * DIM_;
  } else if (row < 2048) {
    src = wb + (size_t)(row - 1024) * DIM_;
    dst = wn + (size_t)row * DIM_;
  } else {
    src = wo + (size_t)(row - 2048) * DIM_;
    dst = won + (size_t)(row - 2048) * DIM_;
  }
  float s = 0.f;
  for (int i = t; i < DIM_; i += 256) {
    float v = src[i];
    s += v * v;
  }
  __shared__ float red[256];
  red[t] = s;
  __syncthreads();
  for (int off = 128; off > 0; off >>= 1) {
    if (t < off) red[t] += red[t + off];
    __syncthreads();
  }
  float norm = sqrtf(red[0]);
  float scale = 1.0f / ((EPS_ + norm * (1.0f / 32.0f)) * 32.0f);
  for (int i = t; i < DIM_; i += 256) dst[i] = src[i] * scale;
}

// =====================================================================
// Kernel 2: fp32 WMMA GEMM with async global->LDS double buffering.
// Out[b][o][l] = sum_i W[o][i] * X[b][i][l]
// W: [M x 1024] row-major (shared across batch), X/Out per batch
// [1024 x L_] / [M x L_] row-major.  Block tile 128x128, K-tile 32.
// 8 wave32 waves: wave grid 4 (M) x 2 (N); each wave 2x4 16x16 subtiles.
// Staging uses GLOBAL_LOAD_ASYNC_TO_LDS_B128 (ASYNCcnt); the next k-tile
// is issued before computing the current one, synced with
// s_wait_asynccnt 8 (async loads complete in order -> <=8 outstanding
// means the current buffer's 8 transfers have landed).
// =====================================================================
__global__ __launch_bounds__(256) void wmma_gemm_kernel(
    const float* __restrict__ W, const float* __restrict__ X,
    float* __restrict__ Out, int M) {
  const int L = L_, K = DIM_;
  const float* Xb = X + (size_t)blockIdx.z * K * L;
  float* Ob = Out + (size_t)blockIdx.z * (size_t)M * L;
  const int om = blockIdx.y * BM;  // output row base
  const int ln = blockIdx.x * BN;  // output col base

  __shared__ float As[2][BM][APAD];
  __shared__ float Bs[2][BK][BPAD];

  const int t = threadIdx.x;
  const int lane = t & 31;
  const int wid = t >> 5;
  const int wm = wid & 3;   // 0..3 -> 32-row strip
  const int wnw = wid >> 2; // 0..1 -> 64-col strip
  const int lr = lane & 15; // row/col within 16x16 frag
  const int hi = lane >> 4; // half-wave select (K pair / M+8)

  // issue one k-tile's staging as 8 async B128 transfers per thread
  auto issue_tile = [&](int buf, int kt) {
#pragma unroll
    for (int i = 0; i < 4; ++i) {
      int v = t + i * 256;
      int r = v >> 3;
      int c = (v & 7) << 2;
      unsigned lds = (unsigned)(size_t)&As[buf][r][c];  // low 32b = LDS addr
      const float* gp = &W[(size_t)(om + r) * K + kt + c];
      asm volatile("global_load_async_to_lds_b128 %0, %1, off"
                   :
                   : "v"(lds), "v"(gp)
                   : "memory");
    }
#pragma unroll
    for (int i = 0; i < 4; ++i) {
      int v = t + i * 256;
      int r = v >> 5;
      int c = (v & 31) << 2;
      unsigned lds = (unsigned)(size_t)&Bs[buf][r][c];
      const float* gp = &Xb[(size_t)(kt + r) * L + ln + c];
      asm volatile("global_load_async_to_lds_b128 %0, %1, off"
                   :
                   : "v"(lds), "v"(gp)
                   : "memory");
    }
  };

  v8f acc[2][4];
  const v8f vzero = {0.f, 0.f, 0.f, 0.f, 0.f, 0.f, 0.f, 0.f};
#pragma unroll
  for (int mt = 0; mt < 2; ++mt)
#pragma unroll
    for (int nt = 0; nt < 4; ++nt) acc[mt][nt] = vzero;

  issue_tile(0, 0);

  for (int kt = 0; kt < K; kt += BK) {
    const int cur = (kt >> 5) & 1;
    const bool more = (kt + BK) < K;

    if (more) issue_tile(cur ^ 1, kt + BK);  // overlap next tile's HBM fetch

    // prefetch the tile after next into GL2 (global_prefetch_b8)
    if (kt + 2 * BK < K) {
      __builtin_prefetch(&W[(size_t)(om + (t >> 1)) * K + kt + 2 * BK], 0, 1);
      __builtin_prefetch(&Xb[(size_t)(kt + 2 * BK + (t >> 3)) * L + ln], 0, 1);
    }

    if (more)
      asm volatile("s_wait_asynccnt 0x8" ::: "memory");
    else
      asm volatile("s_wait_asynccnt 0x0" ::: "memory");
    __syncthreads();

    // ---- 8 k-steps of V_WMMA_F32_16X16X4_F32 ----
#pragma unroll
    for (int kk = 0; kk < BK; kk += 4) {
      v2f a[2], b[4];
#pragma unroll
      for (int mt = 0; mt < 2; ++mt) {
        // A 16x4 frag: lanes 0-15 K={kk,kk+1}, lanes 16-31 K={kk+2,kk+3}
        int rm = wm * 32 + mt * 16 + lr;
        a[mt].x = As[cur][rm][kk + 2 * hi];
        a[mt].y = As[cur][rm][kk + 2 * hi + 1];
      }
#pragma unroll
      for (int nt = 0; nt < 4; ++nt) {
        // B 4x16 frag: VGPR0 -> K=kk+2*hi, VGPR1 -> K=kk+2*hi+1
        int cn = wnw * 64 + nt * 16 + lr;
        b[nt].x = Bs[cur][kk + 2 * hi][cn];
        b[nt].y = Bs[cur][kk + 2 * hi + 1][cn];
      }
#pragma unroll
      for (int mt = 0; mt < 2; ++mt)
#pragma unroll
        for (int nt = 0; nt < 4; ++nt)
          acc[mt][nt] = __builtin_amdgcn_wmma_f32_16x16x4_f32(
              false, a[mt], false, b[nt], (short)0, acc[mt][nt], false, false);
    }
    __syncthreads();  // all waves done reading 'cur' before it is refilled
  }

  // ---- epilogue: D VGPR r -> M=r (lanes 0-15) / M=r+8 (lanes 16-31) ----
#pragma unroll
  for (int mt = 0; mt < 2; ++mt)
#pragma unroll
    for (int nt = 0; nt < 4; ++nt) {
      int gr0 = om + wm * 32 + mt * 16 + hi * 8;
      int gc = ln + wnw * 64 + nt * 16 + lr;
#pragma unroll
      for (int r = 0; r < 8; ++r)
        Ob[(size_t)(gr0 + r) * L + gc] = acc[mt][nt][r];
    }
}

// =====================================================================
// Scan math helpers (numerically stable log-space Heinsen recurrence)
// =====================================================================
__device__ __forceinline__ float softplus_(float x) {
  return fmaxf(x, 0.f) + log1pf(expf(-fabsf(x)));
}
__device__ __forceinline__ float logg_(float x) {
  return (x >= 0.f) ? logf(x + 0.5f) : -softplus_(-x);
}
__device__ __forceinline__ float logaddexp_(float a, float b) {
  float m = fmaxf(a, b);
  float d = fabsf(a - b);
  return m + log1pf(expf(-d));
}

// row decode: row in [0,8192): b = row>>10, dir = (row>>9)&1, h = row&511
__device__ __forceinline__ void row_ptrs(int row, const float* hg,
                                         const float*& hid, const float*& gat,
                                         int& dir) {
  int b = row >> 10;
  int rr = row & 1023;
  dir = rr >> 9;
  int hr = rr & 511;
  hid = hg + ((size_t)b * 2048 + (size_t)dir * 1024 + hr) * L_;
  gat = hg + ((size_t)b * 2048 + (size_t)dir * 1024 + 512 + hr) * L_;
}

// =====================================================================
// Kernel 3: per-(row,chunk) partial scan state: A_c = sum(log_coeffs),
// M_c = logsumexp over chunk of (log_values - local a*).
// =====================================================================
__global__ __launch_bounds__(256) void scan_partial_kernel(
    const float* __restrict__ hg, float* __restrict__ chA,
    float* __restrict__ chM) {
  int gid = blockIdx.x * 256 + threadIdx.x;  // 0 .. 8192*32-1
  int row = gid >> 5;
  int ch = gid & 31;
  const float *hid, *gat;
  int dir;
  row_ptrs(row, hg, hid, gat, dir);
  float A = 0.f, Mv = -INFINITY;
  for (int i = 0; i < CH; ++i) {
    int p = ch * CH + i;
    int l = dir ? (L_ - 1 - p) : p;
    float g = gat[l];
    float hd = hid[l];
    float lc = -softplus_(g);
    float lv = -softplus_(-g) + logg_(hd);
    A += lc;
    Mv = logaddexp_(Mv, lv - A);
  }
  chA[gid] = A;
  chM[gid] = Mv;
}

// =====================================================================
// Kernel 4: per-row sequential combine over 32 chunk summaries ->
// carry-in (A_prev, S_prev) for every chunk.
// =====================================================================
__global__ __launch_bounds__(256) void scan_combine_kernel(
    const float* __restrict__ chA, const float* __restrict__ chM,
    float* __restrict__ carA, float* __restrict__ carS) {
  int row = blockIdx.x * 256 + threadIdx.x;  // 0..8191
  float A = 0.f, S = -INFINITY;
  for (int ch = 0; ch < NCH; ++ch) {
    int idx = row * NCH + ch;
    carA[idx] = A;
    carS[idx] = S;
    S = logaddexp_(S, chM[idx] - A);
    A += chA[idx];
  }
}

// =====================================================================
// Kernel 5: recompute chunk with carry-in, emit h = exp(a* + lse) into
// hcat[B][1024][L] (fore -> rows 0..511, back -> rows 512..1023, with
// back written at original l, i.e. already re-reversed).
// =====================================================================
__global__ __launch_bounds__(256) void scan_final_kernel(
    const float* __restrict__ hg, const float* __restrict__ carA,
    const float* __restrict__ carS, float* __restrict__ hcat) {
  int gid = blockIdx.x * 256 + threadIdx.x;
  int row = gid >> 5;
  int ch = gid & 31;
  const float *hid, *gat;
  int dir;
  row_ptrs(row, hg, hid, gat, dir);
  int b = row >> 10;
  int hr = row & 511;
  float* out = hcat + ((size_t)b * 1024 + (size_t)dir * 512 + hr) * L_;
  float Aprev = carA[gid];
  float S = carS[gid];
  float A = 0.f;
  for (int i = 0; i < CH; ++i) {
    int p = ch * CH + i;
    int l = dir ? (L_ - 1 - p) : p;
    float g = gat[l];
    float hd = hid[l];
    float lc = -softplus_(g);
    float lv = -softplus_(-g) + logg_(hd);
    A += lc;
    S = logaddexp_(S, lv - (Aprev + A));
    out[l] = expf(Aprev + A + S);
  }
}

// =====================================================================
// Host-side launch
// =====================================================================
extern "C" void kernel_launch(void* const* d_in, const int* in_sizes, int n_in,
                              void* d_out, int out_size, void* d_ws,
                              size_t ws_size, hipStream_t stream) {
  const float* x = (const float*)d_in[0];       // [8,1024,4096]
  const float* w_fore = (const float*)d_in[1];  // [1024,1024]
  const float* w_back = (const float*)d_in[2];  // [1024,1024]
  const float* w_out = (const float*)d_in[3];   // [1024,1024]
  float* out = (float*)d_out;                   // [8,1024,4096]

  float* ws = (float*)d_ws;
  float* wn = ws;                                   // 2048*1024 normalized fore+back
  float* won = wn + (size_t)2048 * 1024;            // 1024*1024 normalized w_out
  float* hg = won + (size_t)1024 * 1024;            // [8,2048,4096]
  float* hcat = hg + (size_t)B_ * 2048 * L_;        // [8,1024,4096]
  float* chA = hcat + (size_t)B_ * 1024 * L_;       // [8192,32]
  float* chM = chA + (size_t)NROWS * NCH;
  float* carA = chM + (size_t)NROWS * NCH;
  float* carS = carA + (size_t)NROWS * NCH;

  // 1. weight normalization (3072 rows)
  norm_rows_kernel<<<3072, 256, 0, stream>>>(w_fore, w_back, w_out, wn, won);

  // 2. fused fore+back input projection: [2048x1024] x [1024 x 4096] per batch
  wmma_gemm_kernel<<<dim3(L_ / BN, 2048 / BM, B_), 256, 0, stream>>>(wn, x, hg,
                                                                     2048);

  // 3-5. chunked log-space MinGRU scan (fore + back in one pass)
  scan_partial_kernel<<<(NROWS * NCH) / 256, 256, 0, stream>>>(hg, chA, chM);
  scan_combine_kernel<<<NROWS / 256, 256, 0, stream>>>(chA, chM, carA, carS);
  scan_final_kernel<<<(NROWS * NCH) / 256, 256, 0, stream>>>(hg, carA, carS,
                                                             hcat);

  // 6. output projection: [1024x1024] x [1024 x 4096] per batch
  wmma_gemm_kernel<<<dim3(L_ / BN, 1024 / BM, B_), 256, 0, stream>>>(won, hcat,
                                                                     out, 1024);
}